// cgss_37821482009417
// MI455X (gfx1250) — compile-verified
//
#include <hip/hip_runtime.h>
#include <hip/hip_bf16.h>

typedef float v2f __attribute__((ext_vector_type(2)));
typedef float v8f __attribute__((ext_vector_type(8)));

#define C_DIM 224
#define T_DIM 224
#define LDS_STRIDE 228   // 912 B row stride: 16B-aligned rows for B128 async
#define NBLK 14          // 224 / 16
#define CT (C_DIM * T_DIM)

__device__ __forceinline__ void async_wait0() {
#if __has_builtin(__builtin_amdgcn_s_wait_asynccnt)
    __builtin_amdgcn_s_wait_asynccnt(0);
#else
    asm volatile("s_wait_asynccnt 0" ::: "memory");
#endif
}

// ---------------------------------------------------------------------------
// Tiny kernel: compute the 13 Gaussian taps from scalar w into workspace.
// ---------------------------------------------------------------------------
__global__ void cgss_taps_kernel(const float* __restrict__ w,
                                 float* __restrict__ taps) {
    float sigma = fminf(fmaxf(w[0], 0.4f), 10.0f);
    float c = 0.5f / (sigma * sigma);
    float norm = 0.0f;
    for (int i = 0; i < 130; ++i) {
        float r = -60.0f + (120.0f / 129.0f) * (float)i;
        norm += expf(-r * r * c);
    }
    int i = (int)threadIdx.x;
    if (i < 13) {
        float r = (float)(i - 6);
        taps[i] = expf(-r * r * c) / norm;
    }
}

// ---------------------------------------------------------------------------
// Fused kernel: one block per batch.
//   Phase 1: async global->LDS stage of x[b] (B128, padded row stride)
//   Phase 2: x_mean = banded-Toeplitz A (13-tap) * x via V_WMMA_F32_16X16X4_F32
//            I = min(x, x_mean), written back in place (wave-private columns)
//   Phase 3: LIF scan along T per channel, spikes overwrite I in LDS
//   Phase 4: async LDS->global drain of spikes (B128)
// ---------------------------------------------------------------------------
__global__ __launch_bounds__(256) void cgss_snn_kernel(
        const float* __restrict__ x,
        const float* __restrict__ taps,
        float* __restrict__ out) {
    extern __shared__ float lds[];
    float* xs      = lds;                      // 224 * 228 floats
    float* tapsLds = lds + C_DIM * LDS_STRIDE; // 16 floats

    const int tid = (int)threadIdx.x;
    const int b   = (int)blockIdx.x;
    const size_t base = (size_t)b * CT;

    if (tid < 16) tapsLds[tid] = (tid < 13) ? taps[tid] : 0.0f;

    // ---- Phase 1: async stage input into LDS (CT/4 = 12544 = 49*256) ----
    const float4* xin4 = (const float4*)(x + base);
    for (int i4 = tid; i4 < CT / 4; i4 += 256) {
        int c  = i4 / (T_DIM / 4);
        int tq = i4 % (T_DIM / 4);
        unsigned lds_off = (unsigned)((c * LDS_STRIDE + tq * 4) * sizeof(float));
        unsigned long long gaddr = (unsigned long long)(const void*)(xin4 + i4);
        asm volatile("global_load_async_to_lds_b128 %0, %1, off"
                     :: "v"(lds_off), "v"(gaddr) : "memory");
    }
    async_wait0();
    __syncthreads();

    // ---- Phase 2: banded conv along C via WMMA ----
    const int wave = tid >> 5;
    const int lane = tid & 31;
    const int m    = lane & 15;        // row (A) / col (B,C,D) within tile
    const int half = (lane >> 4) & 1;  // lane group
    const int koff = half * 2;         // K offset carried by lane group

    // A operands (Toeplitz => same for every row-block): role 0 = left
    // neighbor block, 1 = diagonal, 2 = right neighbor. 4 K-chunks each.
    v2f A[3][4];
#pragma unroll
    for (int ro = 0; ro < 3; ++ro) {
#pragma unroll
        for (int q = 0; q < 4; ++q) {
            int k  = 4 * q + koff;
            int d0 = k     - m + (ro - 1) * 16 + 6;  // tap index
            int d1 = k + 1 - m + (ro - 1) * 16 + 6;
            A[ro][q].x = (d0 >= 0 && d0 < 13) ? tapsLds[d0] : 0.0f;
            A[ro][q].y = (d1 >= 0 && d1 < 13) ? tapsLds[d1] : 0.0f;
        }
    }

    // Each wave owns whole 16-column panels: tb = wave, wave+8.
    for (int tb = wave; tb < NBLK; tb += 8) {
        const int tc = tb * 16;
        const int n  = lane & 15;

        v2f B0[4], B1[4], B2[4];  // rolling window: blocks cb-1, cb, cb+1
        auto loadB = [&](int nb, v2f* Bq) {
#pragma unroll
            for (int q = 0; q < 4; ++q) {
                int r0 = nb * 16 + 4 * q + koff;
                Bq[q].x = xs[r0       * LDS_STRIDE + tc + n];
                Bq[q].y = xs[(r0 + 1) * LDS_STRIDE + tc + n];
            }
        };
#pragma unroll
        for (int q = 0; q < 4; ++q) { B0[q].x = 0.0f; B0[q].y = 0.0f; }
        loadB(0, B1);
        loadB(1, B2);

        v8f accPrev = {0.f, 0.f, 0.f, 0.f, 0.f, 0.f, 0.f, 0.f};
        for (int cb = 0; cb < NBLK; ++cb) {
            v8f acc = {0.f, 0.f, 0.f, 0.f, 0.f, 0.f, 0.f, 0.f};
#pragma unroll
            for (int q = 0; q < 4; ++q)
                acc = __builtin_amdgcn_wmma_f32_16x16x4_f32(
                    false, A[0][q], false, B0[q], (short)0, acc, false, false);
#pragma unroll
            for (int q = 0; q < 4; ++q)
                acc = __builtin_amdgcn_wmma_f32_16x16x4_f32(
                    false, A[1][q], false, B1[q], (short)0, acc, false, false);
#pragma unroll
            for (int q = 0; q < 4; ++q)
                acc = __builtin_amdgcn_wmma_f32_16x16x4_f32(
                    false, A[2][q], false, B2[q], (short)0, acc, false, false);

            if (cb > 0) {  // write back I = min(x, x_mean) for block cb-1
                int rowBase = (cb - 1) * 16 + half * 8;
#pragma unroll
                for (int i = 0; i < 8; ++i) {
                    int addr = (rowBase + i) * LDS_STRIDE + tc + n;
                    float xv = xs[addr];
                    xs[addr] = fminf(xv, accPrev[i]);
                }
            }
            accPrev = acc;
#pragma unroll
            for (int q = 0; q < 4; ++q) { B0[q] = B1[q]; B1[q] = B2[q]; }
            if (cb + 2 < NBLK) {
                loadB(cb + 2, B2);
            } else {
#pragma unroll
                for (int q = 0; q < 4; ++q) { B2[q].x = 0.0f; B2[q].y = 0.0f; }
            }
        }
        {   // final write-back: block 13
            int rowBase = (NBLK - 1) * 16 + half * 8;
#pragma unroll
            for (int i = 0; i < 8; ++i) {
                int addr = (rowBase + i) * LDS_STRIDE + tc + n;
                float xv = xs[addr];
                xs[addr] = fminf(xv, accPrev[i]);
            }
        }
    }
    __syncthreads();

    // ---- Phase 3: LIF scan along T (sequential), spikes overwrite I ----
    if (tid < C_DIM) {
        float mem = 0.0f;
        const int rowOff = tid * LDS_STRIDE;
#pragma unroll 4
        for (int t = 0; t < T_DIM; ++t) {
            float It    = xs[rowOff + t];
            float reset = (mem > 1.0f) ? 1.0f : 0.0f;
            mem = 0.97f * mem + It - reset;
            xs[rowOff + t] = (mem > 1.0f) ? 1.0f : 0.0f;
        }
    }
    __syncthreads();

    // ---- Phase 4: async drain of spikes, LDS -> global ----
    const float4* out4 = (const float4*)(out + base);
    for (int i4 = tid; i4 < CT / 4; i4 += 256) {
        int c  = i4 / (T_DIM / 4);
        int tq = i4 % (T_DIM / 4);
        unsigned lds_off = (unsigned)((c * LDS_STRIDE + tq * 4) * sizeof(float));
        unsigned long long gaddr = (unsigned long long)(const void*)(out4 + i4);
        asm volatile("global_store_async_from_lds_b128 %0, %1, off"
                     :: "v"(gaddr), "v"(lds_off) : "memory");
    }
    async_wait0();
}

extern "C" void kernel_launch(void* const* d_in, const int* in_sizes, int n_in,
                              void* d_out, int out_size, void* d_ws, size_t ws_size,
                              hipStream_t stream) {
    const float* inp = (const float*)d_in[0];   // (512,1,224,224) f32
    const float* w   = (const float*)d_in[1];   // scalar f32
    float* out  = (float*)d_out;                // (512,1,224,224) f32
    float* taps = (float*)d_ws;                 // 13 floats scratch

    cgss_taps_kernel<<<1, 32, 0, stream>>>(w, taps);

    const size_t lds_bytes = (size_t)(C_DIM * LDS_STRIDE + 16) * sizeof(float);
    cgss_snn_kernel<<<512, 256, lds_bytes, stream>>>(inp, taps, out);
}